// ComputeAllAtomCoords_38457137168462
// MI455X (gfx1250) — compile-verified
//
#include <hip/hip_runtime.h>
#include <hip/hip_bf16.h>

// ===========================================================================
// ComputeAllAtomCoords — MI455X (gfx1250)
//
// Memory bound: ~0.7 GFLOP vs ~134 MB HBM -> ~5.7us floor @ 23.3 TB/s.
// 1 thread = 1 residue; per-AA tables (~25 KB) staged to LDS; per-AA rotation
// axes precomputed once per block.
//
// CDNA5 path: the 7 general 4x4 GEMMs per residue (X @ RT_k) are executed on
// the matrix pipe with V_WMMA_F32_16X16X4_F32, batching 4 residues per wave
// into the diagonal blocks of D = A(16x4) @ B(4x16):
//   A row m (lane m / m+16, VGPRs K0..1/K2..3) = X_{m>>2}[m&3][k]   (from LDS)
//   B col n (lane n / n+16)                    = RT_{n>>2}[k][n&3]  (from LDS)
//   diag block g of D = X_g @ RT_g  (full fp32, RTE)
// Operands are fetched straight from LDS with per-lane addressing -> no
// cross-lane register shuffles; results scatter back to LDS (diagonal lanes
// only) and owner threads read them back. All traffic is intra-wave, so LDS
// in-order guarantees make it barrier-free.
// Layout assumption (undocumented in ISA dump): B 4x16 mirrors the A/C split,
// VGPR0 = rows {0 | 2}, VGPR1 = rows {1 | 3} for lanes {0-15 | 16-31}.
// ===========================================================================

#define NAA 22
#define NAT 27
#define NFR 9
#define THREADS 256

typedef __attribute__((ext_vector_type(2))) float v2f;
typedef __attribute__((ext_vector_type(8))) float v8f;

#if defined(__AMDGCN__) && __has_builtin(__builtin_amdgcn_wmma_f32_16x16x4_f32)
#define USE_WMMA 1
#else
#define USE_WMMA 0
#endif

// ---------------------------------------------------------------------------
// small helpers (row-major float[16])
// ---------------------------------------------------------------------------
__device__ __forceinline__ void mm4(float* __restrict__ o,
                                    const float* __restrict__ A,
                                    const float* B) {
#pragma unroll
  for (int i = 0; i < 4; ++i) {
#pragma unroll
    for (int j = 0; j < 4; ++j) {
      o[i * 4 + j] = A[i * 4 + 0] * B[0 + j] + A[i * 4 + 1] * B[4 + j] +
                     A[i * 4 + 2] * B[8 + j] + A[i * 4 + 3] * B[12 + j];
    }
  }
}

__device__ __forceinline__ void rotX_right(float* A, float c, float s) {
#pragma unroll
  for (int i = 0; i < 4; ++i) {
    float a1 = A[i * 4 + 1], a2 = A[i * 4 + 2];
    A[i * 4 + 1] = c * a1 + s * a2;
    A[i * 4 + 2] = c * a2 - s * a1;
  }
}

__device__ __forceinline__ void rotZ_right(float* A, float c, float s) {
#pragma unroll
  for (int i = 0; i < 4; ++i) {
    float a0 = A[i * 4 + 0], a1 = A[i * 4 + 1];
    A[i * 4 + 0] = c * a0 + s * a1;
    A[i * 4 + 1] = c * a1 - s * a0;
  }
}

__device__ __forceinline__ void rotAxis_right(float* A, float ct, float st,
                                              float u0, float u1, float u2) {
  float omc = 1.0f - ct;
  float r00 = ct + u0 * u0 * omc, r01 = u0 * u1 * omc - u2 * st, r02 = u0 * u2 * omc + u1 * st;
  float r10 = u0 * u1 * omc + u2 * st, r11 = ct + u1 * u1 * omc, r12 = u1 * u2 * omc - u0 * st;
  float r20 = u0 * u2 * omc - u1 * st, r21 = u1 * u2 * omc + u0 * st, r22 = ct + u2 * u2 * omc;
#pragma unroll
  for (int i = 0; i < 4; ++i) {
    float a0 = A[i * 4 + 0], a1 = A[i * 4 + 1], a2 = A[i * 4 + 2];
    A[i * 4 + 0] = a0 * r00 + a1 * r10 + a2 * r20;
    A[i * 4 + 1] = a0 * r01 + a1 * r11 + a2 * r21;
    A[i * 4 + 2] = a0 * r02 + a1 * r12 + a2 * r22;
  }
}

__device__ __forceinline__ void ang_cs(float a0, float a1, float& c, float& s) {
  float n = sqrtf(a0 * a0 + a1 * a1) + 1e-6f;  // EPS = 1e-6 (reference)
  c = a0 / n;
  s = a1 / n;
}

__device__ __forceinline__ void store16(float* p, const float* m) {
#pragma unroll
  for (int i = 0; i < 4; ++i)
    ((float4*)p)[i] = make_float4(m[4 * i], m[4 * i + 1], m[4 * i + 2], m[4 * i + 3]);
}

__device__ __forceinline__ void load16(float* m, const float* p) {
#pragma unroll
  for (int i = 0; i < 4; ++i) {
    float4 v = ((const float4*)p)[i];
    m[4 * i + 0] = v.x; m[4 * i + 1] = v.y; m[4 * i + 2] = v.z; m[4 * i + 3] = v.w;
  }
}

// ---------------------------------------------------------------------------
// shared staging: per-AA tables + per-AA rotation axes
// ---------------------------------------------------------------------------
__device__ __forceinline__ void stage_tables(
    int t, const int* __restrict__ base_idx, const float* __restrict__ RTs,
    const float* __restrict__ bxyz, const int* __restrict__ amask, float* sRT,
    float* sBX, int* sIDX, int* sMSK, float* sAX1, float* sAX2) {
  for (int i = t; i < NAA * 7 * 16; i += THREADS) sRT[i] = RTs[i];
  for (int i = t; i < NAA * NAT * 4; i += THREADS) sBX[i] = bxyz[i];
  for (int i = t; i < NAA * NAT; i += THREADS) {
    sIDX[i] = base_idx[i];
    sMSK[i] = amask[i];
  }
  __syncthreads();
  if (t < NAA) {
    const float* bx = &sBX[t * NAT * 4];
    float N0 = bx[0], N1 = bx[1], N2 = bx[2];
    float A0 = bx[4], A1 = bx[5], A2 = bx[6];
    float C0 = bx[8], C1 = bx[9], C2 = bx[10];
    float B0 = bx[16], B1 = bx[17], B2 = bx[18];
    float NCr0 = 0.5f * (C0 + N0), NCr1 = 0.5f * (C1 + N1), NCr2 = 0.5f * (C2 + N2);
    float v0 = B0 - A0, v1 = B1 - A1, v2 = B2 - A2;        // CBr - CAr
    float w0 = NCr0 - A0, w1 = NCr1 - A1, w2 = NCr2 - A2;  // NCr - CAr
    float x0 = v1 * w2 - v2 * w1, x1 = v2 * w0 - v0 * w2, x2 = v0 * w1 - v1 * w0;
    float n = sqrtf(x0 * x0 + x1 * x1 + x2 * x2) + 1e-8f;
    sAX1[t * 3 + 0] = x0 / n; sAX1[t * 3 + 1] = x1 / n; sAX1[t * 3 + 2] = x2 / n;
    float p0 = C0 - N0, p1 = C1 - N1, p2 = C2 - N2;        // NCp
    float q = (p0 * NCr0 + p1 * NCr1 + p2 * NCr2) /
              (NCr0 * NCr0 + NCr1 * NCr1 + NCr2 * NCr2);
    float pp0 = p0 - q * NCr0, pp1 = p1 - q * NCr1, pp2 = p2 - q * NCr2;
    float y0 = v1 * pp2 - v2 * pp1, y1 = v2 * pp0 - v0 * pp2, y2 = v0 * pp1 - v1 * pp0;
    float m = sqrtf(y0 * y0 + y1 * y1 + y2 * y2) + 1e-8f;
    sAX2[t * 3 + 0] = y0 / m; sAX2[t * 3 + 1] = y1 / m; sAX2[t * 3 + 2] = y2 / m;
  }
  __syncthreads();
}

// ---------------------------------------------------------------------------
// RTF0 from N, Ca, C
// ---------------------------------------------------------------------------
__device__ __forceinline__ void computeF0(const float* __restrict__ x, float* F0) {
  float Nx = x[0], Ny = x[1], Nz = x[2];
  float CAx = x[3], CAy = x[4], CAz = x[5];
  float Cx = x[6], Cy = x[7], Cz = x[8];
  float v1x = Cx - CAx, v1y = Cy - CAy, v1z = Cz - CAz;
  float v2x = Nx - CAx, v2y = Ny - CAy, v2z = Nz - CAz;
  float n1 = sqrtf(v1x * v1x + v1y * v1y + v1z * v1z) + 1e-8f;
  float e1x = v1x / n1, e1y = v1y / n1, e1z = v1z / n1;
  float d = e1x * v2x + e1y * v2y + e1z * v2z;
  float u2x = v2x - e1x * d, u2y = v2y - e1y * d, u2z = v2z - e1z * d;
  float n2 = sqrtf(u2x * u2x + u2y * u2y + u2z * u2z) + 1e-8f;
  float e2x = u2x / n2, e2y = u2y / n2, e2z = u2z / n2;
  float e3x = e1y * e2z - e1z * e2y;
  float e3y = e1z * e2x - e1x * e2z;
  float e3z = e1x * e2y - e1y * e2x;
  F0[0] = e1x; F0[1] = e2x; F0[2] = e3x; F0[3] = CAx;
  F0[4] = e1y; F0[5] = e2y; F0[6] = e3y; F0[7] = CAy;
  F0[8] = e1z; F0[9] = e2z; F0[10] = e3z; F0[11] = CAz;
  F0[12] = 0.f; F0[13] = 0.f; F0[14] = 0.f; F0[15] = 1.f;
}

// ---------------------------------------------------------------------------
// atom transform epilogue (frames re-read from global, hits L0/L2)
// ---------------------------------------------------------------------------
__device__ __forceinline__ void atoms_phase(int s, const float* __restrict__ fout,
                                            const float* sBX, const int* sIDX,
                                            const int* sMSK, float* __restrict__ op) {
  const float* bx = &sBX[s * NAT * 4];
  const int* bidx = &sIDX[s * NAT];
  const int* bm = &sMSK[s * NAT];
  const float NANF = __builtin_nanf("");
#pragma unroll 3
  for (int at = 0; at < NAT; ++at) {
    int fi = bidx[at];
    const float4* fr = (const float4*)(fout + fi * 16);
    float4 r0 = fr[0], r1 = fr[1], r2 = fr[2];
    float vx = bx[at * 4 + 0], vy = bx[at * 4 + 1], vz = bx[at * 4 + 2], vw = bx[at * 4 + 3];
    float ox = r0.x * vx + r0.y * vy + r0.z * vz + r0.w * vw;
    float oy = r1.x * vx + r1.y * vy + r1.z * vz + r1.w * vw;
    float oz = r2.x * vx + r2.y * vy + r2.z * vz + r2.w * vw;
    bool m = bm[at] != 0;
    op[at * 3 + 0] = m ? ox : NANF;
    op[at * 3 + 1] = m ? oy : NANF;
    op[at * 3 + 2] = m ? oz : NANF;
  }
}

#if USE_WMMA
// ---------------------------------------------------------------------------
// Wave-cooperative batched 4x4 GEMM: for the 4-residue groups of this wave,
// sOut[r] = sX[r] @ RT_{seq[r]}[rtk].  Operands come straight from LDS with
// per-lane addressing per the V_WMMA_F32_16X16X4_F32 layouts; the diagonal
// blocks of D are scattered back to LDS by the 16 lanes that hold them.
// Intra-wave only -> LDS in-order rules make this barrier-free.
// EXEC must be all ones (caller guarantees a full block, uniform control).
// ---------------------------------------------------------------------------
__device__ __forceinline__ void wave_mm4_rt(float* sX, float* sOut,
                                            const float* sRT, const int* sOFF,
                                            int rtk, int t) {
  const int lane = t & 31;
  const int wbase = t & ~31;        // first residue slot of this wave
  const int m = lane & 15;          // A row index == B column index
  const int g = m >> 2;             // 4x4 block group within the 16x16 tile
  const int c = m & 3;              // local row (A) / local col (B)
  const bool hiHalf = lane >= 16;
  const int k0 = hiHalf ? 2 : 0;    // K elements held by this lane half
  const bool useful = hiHalf ? (g >= 2) : (g < 2);  // lanes holding diag rows
  const int jsel = hiHalf ? (g - 2) : g;            // 0 -> D[0..3], 1 -> D[4..7]
#pragma unroll
  for (int it = 0; it < 8; ++it) {
    const int r = wbase + it * 4 + g;    // residue for this row/col block
    v2f A;
    A.x = sX[r * 16 + c * 4 + k0];       // X_r[c][k0]
    A.y = sX[r * 16 + c * 4 + k0 + 1];   // X_r[c][k0+1]
    const int off = sOFF[r] + rtk * 16;  // RT_{seq[r]}[rtk]
    v2f B;
    B.x = sRT[off + k0 * 4 + c];         // RT[k0][c]
    B.y = sRT[off + (k0 + 1) * 4 + c];   // RT[k0+1][c]
    v8f Cz = {0.f, 0.f, 0.f, 0.f, 0.f, 0.f, 0.f, 0.f};
    v8f D = __builtin_amdgcn_wmma_f32_16x16x4_f32(false, A, false, B,
                                                  (short)0, Cz, false, false);
    if (useful) {  // this lane's column c of residue r's 4x4 result
      float d0 = jsel ? D[4] : D[0];
      float d1 = jsel ? D[5] : D[1];
      float d2 = jsel ? D[6] : D[2];
      float d3 = jsel ? D[7] : D[3];
      sOut[r * 16 + 0 * 4 + c] = d0;
      sOut[r * 16 + 1 * 4 + c] = d1;
      sOut[r * 16 + 2 * 4 + c] = d2;
      sOut[r * 16 + 3 * 4 + c] = d3;
    }
  }
}
#endif  // USE_WMMA

// ---------------------------------------------------------------------------
// full-block kernel (all 256 threads valid -> EXEC all ones for WMMA)
// ---------------------------------------------------------------------------
__global__ __launch_bounds__(THREADS) void ComputeAllAtomCoords_main_kernel(
    const int* __restrict__ seq, const float* __restrict__ xyz,
    const float* __restrict__ alphas, const int* __restrict__ base_idx,
    const float* __restrict__ RTs, const float* __restrict__ bxyz,
    const int* __restrict__ amask, float* __restrict__ out_frames,
    float* __restrict__ out_xyz) {
  __shared__ __attribute__((aligned(16))) float sRT[NAA * 7 * 16];
  __shared__ __attribute__((aligned(16))) float sBX[NAA * NAT * 4];
  __shared__ int sIDX[NAA * NAT];
  __shared__ int sMSK[NAA * NAT];
  __shared__ float sAX1[NAA * 3];
  __shared__ float sAX2[NAA * 3];
#if USE_WMMA
  __shared__ int sOFF[THREADS];
  __shared__ __attribute__((aligned(16))) float sX[THREADS * 16];
  __shared__ __attribute__((aligned(16))) float sOut[THREADS * 16];
#endif

  const int t = threadIdx.x;
  const int idx = blockIdx.x * THREADS + t;

  __builtin_prefetch(&xyz[(size_t)idx * 9], 0, 1);       // global_prefetch_b8
  __builtin_prefetch(&alphas[(size_t)idx * 20], 0, 1);

  stage_tables(t, base_idx, RTs, bxyz, amask, sRT, sBX, sIDX, sMSK, sAX1, sAX2);

  float F0[16];
  computeF0(&xyz[(size_t)idx * 9], F0);

  const int s = seq[idx];
  const float* RT = &sRT[s * 7 * 16];
#if USE_WMMA
  sOFF[t] = s * 7 * 16;  // per-residue RT table offset for B-operand fetches
#endif

  __attribute__((aligned(16))) float a[20];
  const float4* alv = (const float4*)(alphas + (size_t)idx * 20);
#pragma unroll
  for (int i = 0; i < 5; ++i) ((float4*)a)[i] = alv[i];

  float c0, s0, c1, s1, c2, s2, c3, s3, c4, s4, c5, s5, c6, s6, c7, s7, c8, s8, c9, s9;
  ang_cs(a[0], a[1], c0, s0);   ang_cs(a[2], a[3], c1, s1);
  ang_cs(a[4], a[5], c2, s2);   ang_cs(a[6], a[7], c3, s3);
  ang_cs(a[8], a[9], c4, s4);   ang_cs(a[10], a[11], c5, s5);
  ang_cs(a[12], a[13], c6, s6); ang_cs(a[14], a[15], c7, s7);
  ang_cs(a[16], a[17], c8, s8); ang_cs(a[18], a[19], c9, s9);

  float* fout = out_frames + (size_t)idx * NFR * 16;
  store16(fout + 0 * 16, F0);  // slot 0 = RTF0

#if USE_WMMA
#define PROD_RT(dst, Xsrc, k)                         \
  do {                                                \
    store16(&sX[t * 16], (Xsrc));                     \
    wave_mm4_rt(sX, sOut, sRT, sOFF, (k), t);         \
    load16((dst), &sOut[t * 16]);                     \
  } while (0)
#else
#define PROD_RT(dst, Xsrc, k) mm4((dst), (Xsrc), RT + (k) * 16)
#endif

  float T[16], U[16];
  PROD_RT(T, F0, 0); rotX_right(T, c0, s0); store16(fout + 1 * 16, T);  // RTF1
  PROD_RT(T, F0, 1); rotX_right(T, c1, s1); store16(fout + 2 * 16, T);  // RTF2
  PROD_RT(T, F0, 2); rotX_right(T, c2, s2); store16(fout + 3 * 16, T);  // RTF3

  // RTF8 = RTF0 @ rot_axis(a7, ax1) @ rot_axis(a8, ax2)  (structured -> VALU)
  float F8[16];
#pragma unroll
  for (int i = 0; i < 16; ++i) F8[i] = F0[i];
  rotAxis_right(F8, c7, s7, sAX1[s * 3 + 0], sAX1[s * 3 + 1], sAX1[s * 3 + 2]);
  rotAxis_right(F8, c8, s8, sAX2[s * 3 + 0], sAX2[s * 3 + 1], sAX2[s * 3 + 2]);
  store16(fout + 8 * 16, F8);  // slot 8 = RTF8

  PROD_RT(T, F8, 3); rotX_right(T, c3, s3); rotZ_right(T, c9, s9);
  store16(fout + 4 * 16, T);                                            // RTF4
  PROD_RT(U, T, 4); rotX_right(U, c4, s4); store16(fout + 5 * 16, U);   // RTF5
  PROD_RT(T, U, 5); rotX_right(T, c5, s5); store16(fout + 6 * 16, T);   // RTF6
  PROD_RT(U, T, 6); rotX_right(U, c6, s6); store16(fout + 7 * 16, U);   // RTF7
#undef PROD_RT

  atoms_phase(s, fout, sBX, sIDX, sMSK, out_xyz + (size_t)idx * NAT * 3);
}

// ---------------------------------------------------------------------------
// tail kernel (partial block): pure-VALU path, guarded
// ---------------------------------------------------------------------------
__global__ __launch_bounds__(THREADS) void ComputeAllAtomCoords_tail_kernel(
    const int* __restrict__ seq, const float* __restrict__ xyz,
    const float* __restrict__ alphas, const int* __restrict__ base_idx,
    const float* __restrict__ RTs, const float* __restrict__ bxyz,
    const int* __restrict__ amask, float* __restrict__ out_frames,
    float* __restrict__ out_xyz, int base, int total) {
  __shared__ __attribute__((aligned(16))) float sRT[NAA * 7 * 16];
  __shared__ __attribute__((aligned(16))) float sBX[NAA * NAT * 4];
  __shared__ int sIDX[NAA * NAT];
  __shared__ int sMSK[NAA * NAT];
  __shared__ float sAX1[NAA * 3];
  __shared__ float sAX2[NAA * 3];

  const int t = threadIdx.x;
  const int idx = base + t;
  stage_tables(t, base_idx, RTs, bxyz, amask, sRT, sBX, sIDX, sMSK, sAX1, sAX2);
  if (idx >= total) return;

  float F0[16];
  computeF0(&xyz[(size_t)idx * 9], F0);
  const int s = seq[idx];
  const float* RT = &sRT[s * 7 * 16];

  __attribute__((aligned(16))) float a[20];
  const float4* alv = (const float4*)(alphas + (size_t)idx * 20);
#pragma unroll
  for (int i = 0; i < 5; ++i) ((float4*)a)[i] = alv[i];
  float c0, s0, c1, s1, c2, s2, c3, s3, c4, s4, c5, s5, c6, s6, c7, s7, c8, s8, c9, s9;
  ang_cs(a[0], a[1], c0, s0);   ang_cs(a[2], a[3], c1, s1);
  ang_cs(a[4], a[5], c2, s2);   ang_cs(a[6], a[7], c3, s3);
  ang_cs(a[8], a[9], c4, s4);   ang_cs(a[10], a[11], c5, s5);
  ang_cs(a[12], a[13], c6, s6); ang_cs(a[14], a[15], c7, s7);
  ang_cs(a[16], a[17], c8, s8); ang_cs(a[18], a[19], c9, s9);

  float* fout = out_frames + (size_t)idx * NFR * 16;
  store16(fout + 0 * 16, F0);
  float T[16], U[16];
  mm4(T, F0, RT + 0 * 16); rotX_right(T, c0, s0); store16(fout + 1 * 16, T);
  mm4(T, F0, RT + 1 * 16); rotX_right(T, c1, s1); store16(fout + 2 * 16, T);
  mm4(T, F0, RT + 2 * 16); rotX_right(T, c2, s2); store16(fout + 3 * 16, T);
  float F8[16];
#pragma unroll
  for (int i = 0; i < 16; ++i) F8[i] = F0[i];
  rotAxis_right(F8, c7, s7, sAX1[s * 3 + 0], sAX1[s * 3 + 1], sAX1[s * 3 + 2]);
  rotAxis_right(F8, c8, s8, sAX2[s * 3 + 0], sAX2[s * 3 + 1], sAX2[s * 3 + 2]);
  store16(fout + 8 * 16, F8);
  mm4(T, F8, RT + 3 * 16); rotX_right(T, c3, s3); rotZ_right(T, c9, s9);
  store16(fout + 4 * 16, T);
  mm4(U, T, RT + 4 * 16); rotX_right(U, c4, s4); store16(fout + 5 * 16, U);
  mm4(T, U, RT + 5 * 16); rotX_right(T, c5, s5); store16(fout + 6 * 16, T);
  mm4(U, T, RT + 6 * 16); rotX_right(U, c6, s6); store16(fout + 7 * 16, U);

  atoms_phase(s, fout, sBX, sIDX, sMSK, out_xyz + (size_t)idx * NAT * 3);
}

// ---------------------------------------------------------------------------
extern "C" void kernel_launch(void* const* d_in, const int* in_sizes, int n_in,
                              void* d_out, int out_size, void* d_ws, size_t ws_size,
                              hipStream_t stream) {
  const int* seq      = (const int*)d_in[0];     // (B,L) int32
  const float* xyz    = (const float*)d_in[1];   // (B,L,3,3) f32
  const float* alphas = (const float*)d_in[2];   // (B,L,10,2) f32
  const int* base_idx = (const int*)d_in[3];     // (22,27) int32
  const float* RTs    = (const float*)d_in[4];   // (22,7,4,4) f32
  const float* bxyz   = (const float*)d_in[5];   // (22,27,4) f32
  const int* amask    = (const int*)d_in[6];     // (22,27)

  const int total = in_sizes[0];                 // B*L residues
  float* out_frames = (float*)d_out;                          // total*9*16
  float* out_xyz    = out_frames + (size_t)total * NFR * 16;  // total*27*3

  const int fullBlocks = total / THREADS;        // 512 for B=32, L=4096
  const int tail = total - fullBlocks * THREADS; // 0 for this shape
  if (fullBlocks > 0) {
    ComputeAllAtomCoords_main_kernel<<<fullBlocks, THREADS, 0, stream>>>(
        seq, xyz, alphas, base_idx, RTs, bxyz, amask, out_frames, out_xyz);
  }
  if (tail > 0) {
    ComputeAllAtomCoords_tail_kernel<<<1, THREADS, 0, stream>>>(
        seq, xyz, alphas, base_idx, RTs, bxyz, amask, out_frames, out_xyz,
        fullBlocks * THREADS, total);
  }
}